// GCNEncoder_55662776156338
// MI455X (gfx1250) — compile-verified
//
#include <hip/hip_runtime.h>

typedef __attribute__((ext_vector_type(2))) float v2f;
typedef __attribute__((ext_vector_type(8))) float v8f;

// ---------------- degree / symmetric normalization ----------------
__global__ void k_fill_ones(float* __restrict__ deg, int n) {
  int i = blockIdx.x * blockDim.x + threadIdx.x;
  if (i < n) deg[i] = 1.0f;                       // self-loop counts as 1
}

__global__ void k_count_deg(const int* __restrict__ dst, float* __restrict__ deg, int E) {
  int e = blockIdx.x * blockDim.x + threadIdx.x;
  if (e < E) atomicAdd(&deg[dst[e]], 1.0f);
}

__global__ void k_rsqrt(float* __restrict__ deg, int n) {
  int i = blockIdx.x * blockDim.x + threadIdx.x;
  if (i < n) deg[i] = rsqrtf(deg[i]);             // dis = deg^{-1/2}
}

// ---------------- f32 WMMA GEMM: T[N x 64] = X[N x K] @ W[K x 64] ----------------
// W is given as Wa (K x ncA) | Wb (K x (64-ncA)) packed column-wise (for mu|logvar).
// 8 waves per block; wave (rt,ct) computes a 16x16 tile of a 32x64 block.
__global__ __launch_bounds__(256) void k_gemm_wmma(
    const float* __restrict__ X, const float* __restrict__ Wa,
    const float* __restrict__ Wb, int ncA, int K, int Nrows,
    float* __restrict__ T)
{
  __shared__ float ldsW[128 * 66];   // padded stride 66: avoids stride-64 bank conflicts
  __shared__ float ldsX[32 * 130];   // padded stride K+2

  const int tid = threadIdx.x;
  const int xs  = K + 2;

  // cooperative load of W (K x 64) into LDS
  for (int e = tid; e < K * 64; e += 256) {
    int k = e >> 6, n = e & 63;
    ldsW[k * 66 + n] = (n < ncA) ? Wa[k * ncA + n] : Wb[k * (64 - ncA) + (n - ncA)];
  }
  // cooperative load of 32-row X tile into LDS (row-coalesced)
  const long row0 = (long)blockIdx.x * 32;
  for (int e = tid; e < 32 * K; e += 256) {
    int r = e / K, c = e - r * K;
    long gr = row0 + r; if (gr >= Nrows) gr = Nrows - 1;   // clamp tail reads
    ldsX[r * xs + c] = X[gr * K + c];
  }
  __syncthreads();

  const int wave = tid >> 5, lane = tid & 31;
  const int rt = wave >> 2, ct = wave & 3;       // 2 row-tiles x 4 col-tiles
  const int l16  = lane & 15;
  const int koff = (lane >> 4) << 1;             // ISA: lanes 16-31 hold K=2,3

  const float* ax = &ldsX[(rt * 16 + l16) * xs];
  const float* bw = &ldsW[ct * 16 + l16];

  v8f acc = {0.f, 0.f, 0.f, 0.f, 0.f, 0.f, 0.f, 0.f};
  for (int k = 0; k < K; k += 4) {
    v2f a, b;
    a.x = ax[k + koff];
    a.y = ax[k + koff + 1];
    b.x = bw[(k + koff) * 66];
    b.y = bw[(k + koff + 1) * 66];
    // D(16x16,f32) = A(16x4,f32) x B(4x16,f32) + C
    acc = __builtin_amdgcn_wmma_f32_16x16x4_f32(false, a, false, b,
                                                (short)0, acc, false, false);
  }

  // C/D layout: VGPR r -> M=r (lanes 0-15) or M=8+r (lanes 16-31); N = lane%16
  const long rbase = row0 + rt * 16 + ((lane >> 4) << 3);
  const int  col   = ct * 16 + l16;
  for (int r = 0; r < 8; ++r) {
    long gr = rbase + r;
    if (gr < Nrows) T[gr * 64 + col] = acc[r];
  }
}

// ---------------- aggregation ----------------
// agg[i][:] = T[i][:] * dis[i]^2   (self-loop term; also serves as the zero-init)
__global__ void k_init_agg(const float* __restrict__ T, const float* __restrict__ dis,
                           float* __restrict__ agg, int n2 /* = N*32 float2s */, int ignored) {
  int idx = blockIdx.x * blockDim.x + threadIdx.x;
  if (idx >= n2) return;
  int row = idx >> 5;                    // 32 float2 per 64-wide row
  float s = dis[row]; s = s * s;
  float2 v = ((const float2*)T)[idx];
  float2 o; o.x = v.x * s; o.y = v.y * s;
  ((float2*)agg)[idx] = o;
}

// one wave per edge: gather 64 contiguous floats of T[src], scale, atomically add to agg[dst]
__global__ __launch_bounds__(256) void k_edge_scatter64(
    const float* __restrict__ T, const float* __restrict__ dis,
    const int* __restrict__ src, const int* __restrict__ dst,
    float* __restrict__ agg, int E)
{
  int e    = (blockIdx.x * 256 + threadIdx.x) >> 5;
  int lane = threadIdx.x & 31;
  if (e >= E) return;
  int s = src[e], d = dst[e];
  float norm = dis[s] * dis[d];
  float2 v = ((const float2*)(T + (long)s * 64))[lane];
  float* ap = agg + (long)d * 64 + lane * 2;
  atomicAdd(ap,     v.x * norm);
  atomicAdd(ap + 1, v.y * norm);
}

// h = relu(agg + b)
__global__ void k_finalize_relu(const float* __restrict__ agg, const float* __restrict__ b,
                                float* __restrict__ h, int n64) {
  int i = blockIdx.x * blockDim.x + threadIdx.x;
  if (i >= n64) return;
  float v = agg[i] + b[i & 63];
  h[i] = fmaxf(v, 0.0f);
}

// split packed [mu|logvar] agg (N x 64) into d_out = [mu (N x 32) ; logvar (N x 32)]
__global__ void k_finalize_mulv(const float* __restrict__ agg,
                                const float* __restrict__ bmu, const float* __restrict__ blv,
                                float* __restrict__ out, int n64, int N) {
  int i = blockIdx.x * blockDim.x + threadIdx.x;
  if (i >= n64) return;
  int row = i >> 6, n = i & 63;
  float v = agg[i];
  if (n < 32) out[(long)row * 32 + n]                    = v + bmu[n];
  else        out[(long)N * 32 + (long)row * 32 + n - 32] = v + blv[n - 32];
}

// ---------------- host launcher ----------------
extern "C" void kernel_launch(void* const* d_in, const int* in_sizes, int n_in,
                              void* d_out, int out_size, void* d_ws, size_t ws_size,
                              hipStream_t stream) {
  const float* x   = (const float*)d_in[0];
  const int*   ei  = (const int*)  d_in[1];
  const float* W1  = (const float*)d_in[2];
  const float* b1  = (const float*)d_in[3];
  const float* W2  = (const float*)d_in[4];
  const float* b2  = (const float*)d_in[5];
  const float* Wmu = (const float*)d_in[6];
  const float* bmu = (const float*)d_in[7];
  const float* Wlv = (const float*)d_in[8];
  const float* blv = (const float*)d_in[9];
  float* out = (float*)d_out;

  const int N = in_sizes[0] / 128;
  const int E = in_sizes[1] / 2;
  const int* src = ei;
  const int* dst = ei + E;

  // carve scratch (256B aligned): dis[N], t[N*64], agg[N*64], h[N*64]
  size_t off = 0;
  auto carve = [&](size_t bytes) -> char* {
    char* p = (char*)d_ws + off;
    off = (off + bytes + 255) & ~(size_t)255;
    return p;
  };
  float* dis = (float*)carve((size_t)N * sizeof(float));
  float* t   = (float*)carve((size_t)N * 64 * sizeof(float));
  float* agg = (float*)carve((size_t)N * 64 * sizeof(float));
  float* h   = (float*)carve((size_t)N * 64 * sizeof(float));
  (void)ws_size;

  const int TB = 256;
  const int gN    = (N + TB - 1) / TB;
  const int gE    = (E + TB - 1) / TB;
  const int gE32  = (int)(((long)E * 32 + TB - 1) / TB);
  const int gN32  = (int)(((long)N * 32 + TB - 1) / TB);
  const int gN64  = (int)(((long)N * 64 + TB - 1) / TB);
  const int gGemm = (N + 31) / 32;
  const int n64   = N * 64;

  // normalization
  k_fill_ones<<<gN, TB, 0, stream>>>(dis, N);
  k_count_deg<<<gE, TB, 0, stream>>>(dst, dis, E);
  k_rsqrt   <<<gN, TB, 0, stream>>>(dis, N);

  // layer 1: t = x @ W1 (K=128) ; agg ; h1 = relu(agg + b1)
  k_gemm_wmma     <<<gGemm, TB, 0, stream>>>(x, W1, W1, 64, 128, N, t);
  k_init_agg      <<<gN32,  TB, 0, stream>>>(t, dis, agg, N * 32, 0);
  k_edge_scatter64<<<gE32,  TB, 0, stream>>>(t, dis, src, dst, agg, E);
  k_finalize_relu <<<gN64,  TB, 0, stream>>>(agg, b1, h, n64);

  // layer 2: t = h1 @ W2 (K=64) ; agg ; h2 = relu(agg + b2)  (h overwritten in place-safe order)
  k_gemm_wmma     <<<gGemm, TB, 0, stream>>>(h, W2, W2, 64, 64, N, t);
  k_init_agg      <<<gN32,  TB, 0, stream>>>(t, dis, agg, N * 32, 0);
  k_edge_scatter64<<<gE32,  TB, 0, stream>>>(t, dis, src, dst, agg, E);
  k_finalize_relu <<<gN64,  TB, 0, stream>>>(agg, b2, h, n64);

  // heads: t = h2 @ [Wmu | Wlv] (K=64, packed 64-wide -> single edge pass for both)
  k_gemm_wmma     <<<gGemm, TB, 0, stream>>>(h, Wmu, Wlv, 32, 64, N, t);
  k_init_agg      <<<gN32,  TB, 0, stream>>>(t, dis, agg, N * 32, 0);
  k_edge_scatter64<<<gE32,  TB, 0, stream>>>(t, dis, src, dst, agg, E);
  k_finalize_mulv <<<gN64,  TB, 0, stream>>>(agg, bmu, blv, out, n64, N);
}